// SAGEGNNEncoder_14594298872184
// MI455X (gfx1250) — compile-verified
//
#include <hip/hip_runtime.h>

#define N_NODES 100000
#define N_EDGES 1600000
#define DIM 64

typedef __attribute__((ext_vector_type(2))) float v2f;
typedef __attribute__((ext_vector_type(8))) float v8f;

// Monotone key transform: unsigned compare on keys == float compare on values.
//   f >= 0 : key = bits | 0x80000000
//   f <  0 : key = ~bits
__device__ __forceinline__ unsigned float_to_key(float f) {
  unsigned b = __float_as_uint(f);
  unsigned mask = (unsigned)(((int)b) >> 31) | 0x80000000u;
  return b ^ mask;
}
__device__ __forceinline__ float key_to_float(unsigned k) {
  unsigned mask = (unsigned)(((int)(~k)) >> 31) | 0x80000000u;
  return __uint_as_float(k ^ mask);
}

#define INIT_KEY 0x007FFFFFu  // float_to_key(-inf): unsigned-min of all finite keys

// ---------------------------------------------------------------------------
// 1) Fill the key accumulator with key(-inf)  (uint4 stores, 1.6M threads)
// ---------------------------------------------------------------------------
__global__ void init_agg_kernel(uint4* __restrict__ aggk) {
  int i = blockIdx.x * blockDim.x + threadIdx.x;
  if (i < N_NODES * (DIM / 4)) {
    aggk[i] = make_uint4(INIT_KEY, INIT_KEY, INIT_KEY, INIT_KEY);
  }
}

// ---------------------------------------------------------------------------
// 2) Edge scatter-max.  64 threads per edge -> the edge id (tid>>6) is
//    wave-uniform; readfirstlane forces the src/dst index loads onto the
//    scalar path (s_load_b64).  Each lane then does ONE coalesced b32 load
//    of the source row and ONE global_atomic_max_u32 on the monotone key
//    into the L2-resident (25.6MB << 192MB) accumulator.  Branch-free,
//    order-independent -> deterministic.
// ---------------------------------------------------------------------------
__global__ void scatter_max_kernel(const long long* __restrict__ src,
                                   const long long* __restrict__ dst,
                                   const float* __restrict__ x,
                                   unsigned* __restrict__ aggk) {
  int tid = blockIdx.x * blockDim.x + threadIdx.x;
  int e = __builtin_amdgcn_readfirstlane(tid >> 6);  // wave-uniform edge id
  int f = tid & (DIM - 1);                           // feature id
  if (e < N_EDGES) {
    int s = (int)src[e];  // scalar loads (uniform address)
    int d = (int)dst[e];
    unsigned key = float_to_key(x[s * DIM + f]);
    atomicMax(&aggk[d * DIM + f], key);              // global_atomic_max_u32
  }
}

// ---------------------------------------------------------------------------
// 3) Combine: out = relu( fix(untransform(aggk)) @ Wl + bl + x @ Wr )
//    One wave32 per 16x16 output tile, K=64 -> 16 k-steps, two
//    v_wmma_f32_16x16x4_f32 per step accumulating into a shared C.
//    Fragment layouts per CDNA5 ISA 7.12.2 (f32 A 16x4 / B 4x16 / C 16x16).
// ---------------------------------------------------------------------------
__global__ void __launch_bounds__(256)
sage_combine_kernel(const unsigned* __restrict__ aggk,
                    const float* __restrict__ xin,
                    const float* __restrict__ Wl,
                    const float* __restrict__ bl,
                    const float* __restrict__ Wr,
                    float* __restrict__ out) {
  const int wave = (blockIdx.x * blockDim.x + threadIdx.x) >> 5;
  const int lane = threadIdx.x & 31;
  const int m0 = (wave >> 2) * 16;  // node-tile  (6250 tiles)
  const int n0 = (wave & 3) * 16;   // feature-tile (4 tiles)
  const int lh = lane & 15;         // lane within half-wave
  const int hi = lane >> 4;         // 0 = lanes 0-15, 1 = lanes 16-31

  v8f c = {};
  const int arow = (m0 + lh) * DIM;
#pragma unroll
  for (int k0 = 0; k0 < DIM; k0 += 4) {
    const int kb = k0 + 2 * hi;  // A: v0=K0(lo)/K2(hi), v1=K1(lo)/K3(hi); B mirrors
    // --- agg @ Wl: untransform key, isolated-node -inf -> 0 ---
    float a0 = key_to_float(aggk[arow + kb]);
    float a1 = key_to_float(aggk[arow + kb + 1]);
    a0 = (a0 == -__builtin_inff()) ? 0.0f : a0;
    a1 = (a1 == -__builtin_inff()) ? 0.0f : a1;
    v2f a = {a0, a1};
    v2f b = {Wl[kb * DIM + n0 + lh], Wl[(kb + 1) * DIM + n0 + lh]};
    c = __builtin_amdgcn_wmma_f32_16x16x4_f32(false, a, false, b,
                                              (short)0, c, false, false);
    // --- x @ Wr ---
    v2f a2 = {xin[arow + kb], xin[arow + kb + 1]};
    v2f b2 = {Wr[kb * DIM + n0 + lh], Wr[(kb + 1) * DIM + n0 + lh]};
    c = __builtin_amdgcn_wmma_f32_16x16x4_f32(false, a2, false, b2,
                                              (short)0, c, false, false);
  }

  const float bias = bl[n0 + lh];
#pragma unroll
  for (int r = 0; r < 8; ++r) {   // C: VGPR r = row m0+r (lo) / m0+8+r (hi)
    const int row = m0 + r + 8 * hi;
    const float v = c[r] + bias;
    out[row * DIM + n0 + lh] = fmaxf(v, 0.0f);
  }
}

// ---------------------------------------------------------------------------
// Host driver: three layers, key-accumulator re-initialized each layer;
// hidden states ping-pong through the workspace, final layer writes d_out.
// ---------------------------------------------------------------------------
extern "C" void kernel_launch(void* const* d_in, const int* in_sizes, int n_in,
                              void* d_out, int out_size, void* d_ws, size_t ws_size,
                              hipStream_t stream) {
  const float* x = (const float*)d_in[0];
  const long long* ei = (const long long*)d_in[1];  // int64 [2, E]
  const long long* src = ei;
  const long long* dst = ei + N_EDGES;

  const float* Wl[3] = {(const float*)d_in[2], (const float*)d_in[5], (const float*)d_in[8]};
  const float* bl[3] = {(const float*)d_in[3], (const float*)d_in[6], (const float*)d_in[9]};
  const float* Wr[3] = {(const float*)d_in[4], (const float*)d_in[7], (const float*)d_in[10]};

  unsigned* aggk = (unsigned*)d_ws;                       // 25.6 MB (keys)
  float* h1 = (float*)(aggk + (size_t)N_NODES * DIM);     // 25.6 MB
  float* h2 = h1 + (size_t)N_NODES * DIM;                 // 25.6 MB
  float* outp = (float*)d_out;

  const float* lin[3] = {x, h1, h2};
  float* lout[3] = {h1, h2, outp};

  const dim3 initGrid((N_NODES * (DIM / 4) + 255) / 256);        // 6.4M uints
  const dim3 scatGrid(((unsigned)N_EDGES * DIM + 255) / 256);    // 102.4M threads
  const dim3 gemmGrid((N_NODES / 16) * (DIM / 16) / 8);          // 25000 waves / 8 per block

  for (int L = 0; L < 3; ++L) {
    init_agg_kernel<<<initGrid, 256, 0, stream>>>((uint4*)aggk);
    scatter_max_kernel<<<scatGrid, 256, 0, stream>>>(src, dst, lin[L], aggk);
    sage_combine_kernel<<<gemmGrid, 256, 0, stream>>>(aggk, lin[L], Wl[L], bl[L],
                                                      Wr[L], lout[L]);
  }
}